// Tokenizer_1382979469374
// MI455X (gfx1250) — compile-verified
//
#include <hip/hip_runtime.h>

typedef float v2f __attribute__((ext_vector_type(2)));
typedef float v8f __attribute__((ext_vector_type(8)));

// Problem constants
#define NVOCAB 4096
#define NEMBED 128
#define NZCH   256
#define NB     32
#define NHW    256           // 16*16
#define NPIX   (NB * NHW)    // 8192

// ---------------------------------------------------------------------------
// Kernel 0: codebook squared norms  cnorm[v] = |emb[v]|^2
// ---------------------------------------------------------------------------
__global__ void vq_cnorm_kernel(const float* __restrict__ emb,
                                float* __restrict__ cnorm) {
    int v = blockIdx.x * blockDim.x + threadIdx.x;
    if (v >= NVOCAB) return;
    const float* row = emb + (long)v * NEMBED;
    float s = 0.f;
#pragma unroll 8
    for (int e = 0; e < NEMBED; ++e) s += row[e] * row[e];
    cnorm[v] = s;
}

// ---------------------------------------------------------------------------
// Kernel 1: pre_quant_conv (1x1 conv == GEMM [8192x256]x[256x128] + bias)
// One wave (32 threads) computes a 16-pixel x 16-embed tile via 64x
// v_wmma_f32_16x16x4_f32 over K=256.
// z_e layout (b,c,h,w): elem = b*65536 + c*256 + hw
// z   layout (b,e,h,w): elem = b*32768 + e*256 + hw
// ---------------------------------------------------------------------------
__global__ void vq_pre_gemm_kernel(const float* __restrict__ z_e,
                                   const float* __restrict__ pre_w,
                                   const float* __restrict__ pre_b,
                                   float* __restrict__ z_out) {
    const int tile = blockIdx.x;      // 0..511 pixel tile
    const int et   = blockIdx.y;      // 0..7   embed tile
    const int lane = threadIdx.x;
    const int lo = lane & 15, hi = lane >> 4;

    // A loads: lane holds A[m=lo][k=2*hi+v]
    const int  p_a    = tile * 16 + lo;
    const long a_base = (long)(p_a >> 8) * 65536 + (p_a & 255);
    // B loads: lane holds B[k=2*hi+v][n=lo], B[c][e] = pre_w[e][c]
    const int  e_n    = et * 16 + lo;
    const float* brow = pre_w + (long)e_n * NZCH + 2 * hi;

    v8f acc = {};
#pragma unroll 4
    for (int kk = 0; kk < NZCH; kk += 4) {
        const int k0 = kk + 2 * hi;
        v2f a, b;
        a.x = z_e[a_base + (long)k0 * NHW];
        a.y = z_e[a_base + (long)(k0 + 1) * NHW];
        b   = *(const v2f*)(brow + kk);
        acc = __builtin_amdgcn_wmma_f32_16x16x4_f32(
                  false, a, false, b, (short)0, acc, false, false);
    }
    const float bias = pre_b[e_n];
#pragma unroll
    for (int r = 0; r < 8; ++r) {
        const int p = tile * 16 + r + 8 * hi;   // C row = r + 8*hi, col = lo
        z_out[(long)(p >> 8) * 32768 + (long)e_n * NHW + (p & 255)] =
            acc[r] + bias;
    }
}

// ---------------------------------------------------------------------------
// Kernel 2: distance GEMM + argmin.
// score(p,v) = |c_v|^2 - 2 * z_p . c_v   (|z_p|^2 dropped: per-pixel const)
// One wave per 16-pixel tile; A-tile preloaded into 64 VGPRs; sweep all 4096
// codes as 128 iterations x 2 code-tiles with TWO independent accumulator
// chains (hides WMMA pipeline latency at ~1 wave/SIMD occupancy).
// Per-lane running min over its column, then 16-lane LDS reduction per pixel.
// ---------------------------------------------------------------------------
__global__ void vq_dist_argmin_kernel(const float* __restrict__ z,   // (b,e,h,w)
                                      const float* __restrict__ emb,
                                      const float* __restrict__ cnorm,
                                      int* __restrict__ tokens) {
    __shared__ float s_best[16][16];
    __shared__ int   s_idx[16][16];

    const int tile = blockIdx.x;      // 0..511 pixel tile
    const int lane = threadIdx.x;
    const int lo = lane & 15, hi = lane >> 4;

    // Preload A (zf rows) once: K=128 -> 32 v2f per lane
    const int  p_a    = tile * 16 + lo;
    const long a_base = (long)(p_a >> 8) * 32768 + (p_a & 255);
    v2f areg[32];
#pragma unroll
    for (int s = 0; s < 32; ++s) {
        const int k0 = s * 4 + 2 * hi;
        areg[s].x = z[a_base + (long)k0 * NHW];
        areg[s].y = z[a_base + (long)(k0 + 1) * NHW];
    }

    float best0[8], best1[8];
    int   bidx0[8], bidx1[8];
#pragma unroll
    for (int r = 0; r < 8; ++r) {
        best0[r] = 3.4e38f; bidx0[r] = 0;
        best1[r] = 3.4e38f; bidx1[r] = 0;
    }

#pragma unroll 1
    for (int nt = 0; nt < NVOCAB / 32; ++nt) {       // 128 iterations
        const int code0 = nt * 32 + lo;              // chain-0 column
        const int code1 = code0 + 16;                // chain-1 column
        const float* brow0 = emb + (long)code0 * NEMBED + 2 * hi;
        const float* brow1 = emb + (long)code1 * NEMBED + 2 * hi;
        v8f acc0 = {}, acc1 = {};
#pragma unroll
        for (int s = 0; s < 32; ++s) {
            const v2f b0 = *(const v2f*)(brow0 + s * 4);
            const v2f b1 = *(const v2f*)(brow1 + s * 4);
            acc0 = __builtin_amdgcn_wmma_f32_16x16x4_f32(
                       false, areg[s], false, b0, (short)0, acc0, false, false);
            acc1 = __builtin_amdgcn_wmma_f32_16x16x4_f32(
                       false, areg[s], false, b1, (short)0, acc1, false, false);
        }
        const float cn0 = cnorm[code0];
        const float cn1 = cnorm[code1];
#pragma unroll
        for (int r = 0; r < 8; ++r) {
            const float sc0 = cn0 - 2.0f * acc0[r];
            const float sc1 = cn1 - 2.0f * acc1[r];
            if (sc0 < best0[r]) { best0[r] = sc0; bidx0[r] = code0; }
            if (sc1 < best1[r]) { best1[r] = sc1; bidx1[r] = code1; }
        }
    }

    // merge the two chains (prefer lower index on exact ties, like argmin)
#pragma unroll
    for (int r = 0; r < 8; ++r) {
        float bs = best0[r]; int bi = bidx0[r];
        if (best1[r] < bs || (best1[r] == bs && bidx1[r] < bi)) {
            bs = best1[r]; bi = bidx1[r];
        }
        s_best[r + 8 * hi][lo] = bs;
        s_idx[r + 8 * hi][lo]  = bi;
    }
    __syncthreads();
    if (lane < 16) {
        float bs = s_best[lane][0];
        int   bi = s_idx[lane][0];
        for (int j = 1; j < 16; ++j) {
            const float v = s_best[lane][j];
            const int   ii = s_idx[lane][j];
            if (v < bs || (v == bs && ii < bi)) { bs = v; bi = ii; }
        }
        tokens[tile * 16 + lane] = bi;
    }
}

// ---------------------------------------------------------------------------
// Kernel 3: z_q gather  (b,e,h,w)
// ---------------------------------------------------------------------------
__global__ void vq_gather_kernel(const float* __restrict__ emb,
                                 const int* __restrict__ tokens,
                                 float* __restrict__ zq_out) {
    const int i = blockIdx.x * blockDim.x + threadIdx.x;
    if (i >= NPIX * NEMBED) return;
    const int b  = i >> 15;
    const int e  = (i >> 8) & 127;
    const int hw = i & 255;
    const int p  = (b << 8) | hw;
    zq_out[i] = emb[(long)tokens[p] * NEMBED + e];
}

// ---------------------------------------------------------------------------
// Kernel 4: post_quant_conv  rec = z_q @ post_w^T + post_b
// GEMM [8192x128]x[128x256]; A gathered from emb via tokens (== z_q rows,
// since the straight-through estimator makes dec_in == z_q in value).
// rec layout (b,c,h,w): elem = b*65536 + c*256 + hw
// ---------------------------------------------------------------------------
__global__ void vq_post_gemm_kernel(const float* __restrict__ emb,
                                    const int* __restrict__ tokens,
                                    const float* __restrict__ post_w,
                                    const float* __restrict__ post_b,
                                    float* __restrict__ rec_out) {
    const int tile = blockIdx.x;      // 0..511 pixel tile
    const int ct   = blockIdx.y;      // 0..15  channel tile
    const int lane = threadIdx.x;
    const int lo = lane & 15, hi = lane >> 4;

    const int tok     = tokens[tile * 16 + lo];           // A row pixel = lo
    const float* arow = emb + (long)tok * NEMBED + 2 * hi;
    const int  c_n    = ct * 16 + lo;                     // B column
    const float* brow = post_w + (long)c_n * NEMBED + 2 * hi;

    v8f acc = {};
#pragma unroll
    for (int s = 0; s < 32; ++s) {
        const v2f a = *(const v2f*)(arow + s * 4);
        const v2f b = *(const v2f*)(brow + s * 4);
        acc = __builtin_amdgcn_wmma_f32_16x16x4_f32(
                  false, a, false, b, (short)0, acc, false, false);
    }
    const float bias = post_b[c_n];
#pragma unroll
    for (int r = 0; r < 8; ++r) {
        const int p = tile * 16 + r + 8 * hi;
        rec_out[(long)(p >> 8) * 65536 + (long)c_n * NHW + (p & 255)] =
            acc[r] + bias;
    }
}

// ---------------------------------------------------------------------------
extern "C" void kernel_launch(void* const* d_in, const int* in_sizes, int n_in,
                              void* d_out, int out_size, void* d_ws, size_t ws_size,
                              hipStream_t stream) {
    const float* z_e    = (const float*)d_in[0];
    const float* pre_w  = (const float*)d_in[1];
    const float* pre_b  = (const float*)d_in[2];
    const float* emb    = (const float*)d_in[3];
    const float* post_w = (const float*)d_in[4];
    const float* post_b = (const float*)d_in[5];

    float* out = (float*)d_out;
    float* z   = out;                       // 32*128*16*16 = 1048576
    float* zq  = out + 1048576;             // 1048576
    float* rec = out + 2097152;             // 32*256*16*16 = 2097152

    float* cnorm  = (float*)d_ws;                                 // 16 KB
    int*   tokens = (int*)((char*)d_ws + NVOCAB * sizeof(float)); // 32 KB

    // 0) codebook norms
    vq_cnorm_kernel<<<NVOCAB / 256, 256, 0, stream>>>(emb, cnorm);

    // 1) pre-quant conv (writes z)
    dim3 g1(NPIX / 16, NEMBED / 16);
    vq_pre_gemm_kernel<<<g1, 32, 0, stream>>>(z_e, pre_w, pre_b, z);

    // 2) distance + argmin (reads z from d_out, writes tokens)
    vq_dist_argmin_kernel<<<NPIX / 16, 32, 0, stream>>>(z, emb, cnorm, tokens);

    // 3) z_q gather
    vq_gather_kernel<<<(NPIX * NEMBED) / 256, 256, 0, stream>>>(emb, tokens, zq);

    // 4) post-quant conv (writes rec)
    dim3 g4(NPIX / 16, NZCH / 16);
    vq_post_gemm_kernel<<<g4, 32, 0, stream>>>(emb, tokens, post_w, post_b, rec);
}